// BraLM_22479858827803
// MI455X (gfx1250) — compile-verified
//
#include <hip/hip_runtime.h>
#include <cstddef>

#define Hh 32
#define Ss 64
#define Kk 64
#define Bb 32

typedef __attribute__((ext_vector_type(2))) float v2f;
typedef __attribute__((ext_vector_type(8))) float v8f;

__device__ __forceinline__ float gelu_exact(float x) {
    return 0.5f * x * (1.0f + erff(x * 0.70710678118654752440f));
}

// pe[p][d]: d even -> sin(p * 10000^(d/32)), d odd -> cos(p * 10000^((d-1)/32))
__device__ __forceinline__ float pe_val(int p, int d) {
    float ex  = (float)(d & ~1) * (1.0f / 32.0f);
    float div = expf(ex * 9.210340371976184f); // 10000^ex
    float ang = (float)p * div;
    return (d & 1) ? cosf(ang) : sinf(ang);
}

// ---------------------------------------------------------------------------
// Phase 1: sequential k=0 chain. Batch-independent -> 32 independent waves.
// Each block (32 threads = 1 wave) owns batch b; lane d owns channel d.
// Emits energies[b][i][0:32] for phase 2.
// ---------------------------------------------------------------------------
__global__ void __launch_bounds__(32)
phase1_chain(const int* __restrict__ neighbor_ids,
             const int* __restrict__ param_indices,
             const float* __restrict__ weights,
             const float* __restrict__ biases,
             const float* __restrict__ node_bias,
             const float* __restrict__ positions,
             float* __restrict__ energies) {
    __shared__ float sm_cache[Ss][Hh];   // nxt[:,0] history for this b
    __shared__ float sm_pos[Ss];

    const int b = blockIdx.x;
    const int d = threadIdx.x;           // 0..31

    if (d * 2 < Ss) { sm_pos[d * 2] = positions[d * 2]; sm_pos[d * 2 + 1] = positions[d * 2 + 1]; }
    __syncthreads();

    // init_energy = gelu(1/H + node_bias[neighbor_ids[b,0,0,0]] + pe[0])
    const int nid = neighbor_ids[(size_t)b * Ss * Kk * 2];
    float e = gelu_exact(1.0f / 32.0f + node_bias[(size_t)nid * Hh + d] + pe_val(0, d));

    for (int i = 0; i < Ss; ++i) {
        if (i > 0) {
            // softmax(positions[:i]) weighted mean of cache rows 0..i-1
            float mx = -1e30f;
            for (int s = 0; s < i; ++s) mx = fmaxf(mx, sm_pos[s]);
            float den = 0.0f, pooled = 0.0f;
            for (int s = 0; s < i; ++s) {
                float w = expf(sm_pos[s] - mx);
                den += w;
                pooled += w * sm_cache[s][d];
            }
            e = pooled / den;
        }
        energies[((size_t)b * Ss + i) * Hh + d] = e;

        // nxt0 = gelu(e @ W[idx0] + bias[idx0] + pe[i+1])
        const int idx0 = param_indices[((size_t)b * Ss + i) * Kk];
        const float* Wp = weights + (size_t)idx0 * (Hh * Hh);
        float acc = 0.0f;
        #pragma unroll
        for (int h = 0; h < Hh; ++h)
            acc = fmaf(__shfl(e, h, 32), Wp[h * Hh + d], acc);  // coalesced over d
        float nxt = gelu_exact(acc + biases[(size_t)idx0 * Hh + d] + pe_val(i + 1, d));
        sm_cache[i][d] = nxt;
        __syncthreads();
    }
}

// ---------------------------------------------------------------------------
// Phase 2: one block per (b,i); 8 waves x 8 k's = all K=64 gathered gemvs.
// Each gemv e(1x32) @ W(32x32) runs on one wave via 16 accumulating
// V_WMMA_F32_16X16X4_F32 ops. The A fragments (broadcast energy chunks)
// are identical for every k, so they are built ONCE before the k loop:
// the steady-state inner loop is pure global-load + wmma.
// ---------------------------------------------------------------------------
__global__ void __launch_bounds__(256)
phase2_gather(const float* __restrict__ energies,
              const int* __restrict__ param_indices,
              const float* __restrict__ weights,
              const float* __restrict__ biases,
              float* __restrict__ partials) {
    __shared__ float sm_e[Hh];
    __shared__ float sm_pe[Hh];
    __shared__ float sm_en[Kk];

    const int bi = blockIdx.x;          // 0 .. B*S-1
    const int b  = bi / Ss;
    const int i  = bi % Ss;
    const int tid  = threadIdx.x;
    const int wave = tid >> 5;
    const int lane = tid & 31;

    if (tid < Hh) {
        sm_e[tid]  = energies[((size_t)b * Ss + i) * Hh + tid];
        sm_pe[tid] = pe_val(i + 1, tid);
    }
    __syncthreads();

    const float ev  = sm_e[lane];       // lane d holds e[d]
    const int   hi2 = (lane >> 4) << 1; // 0 for lanes 0-15, 2 for 16-31
    const int   n   = lane & 15;
    const int   base_pk = ((size_t)b * Ss + i) * Kk;

    // A (16x4 f32) fragments, one per 4-wide reduction chunk j:
    // A[m,c] = e[4j+c] for all m. Layout: lanes 0-15 hold K=0/1 in
    // VGPR0/1, lanes 16-31 hold K=2/3. Invariant across k -> hoisted.
    v2f afrag[8];
    #pragma unroll
    for (int j = 0; j < 8; ++j) {
        afrag[j].x = __shfl(ev, 4 * j + hi2, 32);
        afrag[j].y = __shfl(ev, 4 * j + hi2 + 1, 32);
    }

    const int d0 = n, d1 = 16 + n;
    const float pe0 = sm_pe[d0], pe1 = sm_pe[d1];

    for (int kk = 0; kk < 8; ++kk) {
        const int k   = wave * 8 + kk;
        const int idx = param_indices[base_pk + k];
        const float* Wp = weights + (size_t)idx * (Hh * Hh);

        // Prefetch next gathered matrix (global_prefetch_b8), 32 lanes x 128B.
        if (kk + 1 < 8) {
            const int idxn = param_indices[base_pk + k + 1];
            __builtin_prefetch(weights + (size_t)idxn * (Hh * Hh) + lane * 32, 0, 0);
        }

        // Bias in flight while the WMMA chain runs.
        const float bia0 = biases[(size_t)idx * Hh + d0];
        const float bia1 = biases[(size_t)idx * Hh + d1];

        v8f c0 = {}; v8f c1 = {};
        #pragma unroll
        for (int j = 0; j < 8; ++j) {
            const int r0 = 4 * j + hi2;
            // B (4x16 f32), row-striped: VGPR v holds rows v (lanes 0-15)
            // and v+2 (lanes 16-31).
            v2f b0, b1;
            b0.x = Wp[r0 * Hh + n];            // cols 0..15
            b0.y = Wp[(r0 + 1) * Hh + n];
            b1.x = Wp[r0 * Hh + 16 + n];       // cols 16..31
            b1.y = Wp[(r0 + 1) * Hh + 16 + n];
            c0 = __builtin_amdgcn_wmma_f32_16x16x4_f32(false, afrag[j], false, b0,
                                                       (short)0, c0, false, false);
            c1 = __builtin_amdgcn_wmma_f32_16x16x4_f32(false, afrag[j], false, b1,
                                                       (short)0, c1, false, false);
        }
        // All D rows equal; lane l holds out[n] in c0[0], out[16+n] in c1[0].
        float t0 = gelu_exact(c0[0] + bia0 + pe0);
        float t1 = gelu_exact(c1[0] + bia1 + pe1);
        float sq = t0 * t0 + t1 * t1;
        // reduce over the 16-lane group (halves are duplicates)
        sq += __shfl_xor(sq, 1, 32);
        sq += __shfl_xor(sq, 2, 32);
        sq += __shfl_xor(sq, 4, 32);
        sq += __shfl_xor(sq, 8, 32);
        if (lane == 0) sm_en[k] = sqrtf(sq);
    }
    __syncthreads();

    if (tid == 0) {
        float mx = -1e30f;
        for (int k = 0; k < Kk; ++k) mx = fmaxf(mx, sm_en[k]);
        float s = 0.0f;
        for (int k = 0; k < Kk; ++k) s += expf(sm_en[k] - mx);
        partials[bi] = logf(s) + mx - sm_en[0];
    }
}

// ---------------------------------------------------------------------------
// Phase 3: deterministic fixed-order reduction of B*S=2048 partials.
// ---------------------------------------------------------------------------
__global__ void __launch_bounds__(256)
phase3_reduce(const float* __restrict__ partials, float* __restrict__ out) {
    __shared__ float sm[256];
    const int t = threadIdx.x;
    float s = 0.0f;
    for (int j = 0; j < (Bb * Ss) / 256; ++j) s += partials[t + j * 256];
    sm[t] = s;
    __syncthreads();
    for (int off = 128; off > 0; off >>= 1) {
        if (t < off) sm[t] += sm[t + off];
        __syncthreads();
    }
    if (t == 0) out[0] = sm[0] * (1.0f / ((float)Ss * (float)Bb));
}

extern "C" void kernel_launch(void* const* d_in, const int* in_sizes, int n_in,
                              void* d_out, int out_size, void* d_ws, size_t ws_size,
                              hipStream_t stream) {
    const int*   neighbor_ids  = (const int*)d_in[0];
    const int*   param_indices = (const int*)d_in[1];
    const float* weights       = (const float*)d_in[2];
    const float* biases        = (const float*)d_in[3];
    const float* node_bias     = (const float*)d_in[4];
    const float* positions     = (const float*)d_in[5];

    float* energies = (float*)d_ws;                       // B*S*H floats
    float* partials = energies + (size_t)Bb * Ss * Hh;    // B*S floats

    phase1_chain<<<Bb, 32, 0, stream>>>(neighbor_ids, param_indices, weights,
                                        biases, node_bias, positions, energies);
    phase2_gather<<<Bb * Ss, 256, 0, stream>>>(energies, param_indices, weights,
                                               biases, partials);
    phase3_reduce<<<1, 256, 0, stream>>>(partials, (float*)d_out);
}